// Classifier_4853313045126
// MI455X (gfx1250) — compile-verified
//
#include <hip/hip_runtime.h>
#include <hip/hip_bf16.h>

// ---------------------------------------------------------------------------
// Graph pooling classifier head for MI455X (gfx1250, wave32).
//   pooled = segment_mean(features, batch)           [2048, 256]
//   h = LN(pooled @ W1 + b1) -> LeakyReLU -> @ W2 + b2  -> out [2048]
// Pooling is the bandwidth-dominant stage (512 MB feature read, ~22us at
// 23.3 TB/s) -> b128 streaming loads. Head GEMM: V_WMMA_F32_16X16X4_F32.
// ---------------------------------------------------------------------------

typedef __attribute__((ext_vector_type(2))) float v2f;
typedef __attribute__((ext_vector_type(8))) float v8f;

#define D_IN 256
#define D_H 128
#define NUM_GRAPHS 2048
#define ROWS_PER_BLK 512
#define POOL_STRIDE 260   // 260 % 64 == 4 -> 4-bank skew per row, conflict-free
#define H_STRIDE 132      // 132 % 64 == 4 -> skewed rows for LN readers
#define LN_EPS 1e-5f
#define NEG_SLOPE 0.01f

// ---------------- kernel 0: zero the workspace accumulators ----------------
__global__ void zero_ws_kernel(float* __restrict__ ws, int n) {
    int i = blockIdx.x * blockDim.x + threadIdx.x;
    for (; i < n; i += gridDim.x * blockDim.x) ws[i] = 0.0f;
}

// ---------------- kernel 1: segmented sum (batch is sorted) ----------------
// 256 threads = 4 row-sublanes x 64 column-quads. Each iteration the block
// consumes 4 consecutive rows as fully coalesced global_load_b128. A thread's
// row subsequence (stride 4) is sorted, so register-accumulate per segment
// run and flush with f32 atomics only at run boundaries. One leader thread
// per row-sublane maintains counts (each row counted exactly once).
__global__ void __launch_bounds__(256)
seg_sum_kernel(const float* __restrict__ feat, const int* __restrict__ batch,
               float* __restrict__ sums, float* __restrict__ counts, int n_rows) {
    __shared__ int segS[ROWS_PER_BLK];
    const int t    = threadIdx.x;
    const int q    = t >> 6;          // row sub-lane 0..3
    const int col4 = (t & 63) << 2;   // column base 0,4,...,252
    const long long row0 = (long long)blockIdx.x * ROWS_PER_BLK;
    int nr = (int)((long long)n_rows - row0);
    if (nr > ROWS_PER_BLK) nr = ROWS_PER_BLK;
    if (nr <= 0) return;

    for (int i = t; i < nr; i += 256) segS[i] = batch[row0 + i];
    __syncthreads();

    float4 acc = make_float4(0.f, 0.f, 0.f, 0.f);
    int    cnt = 0;
    for (int ri = q; ri < nr; ri += 4) {
        const float4 v = *(const float4*)&feat[(size_t)(row0 + ri) * D_IN + col4];
        acc.x += v.x; acc.y += v.y; acc.z += v.z; acc.w += v.w;
        ++cnt;
        const int rnext = ri + 4;
        if (rnext >= nr || segS[rnext] != segS[ri]) {
            const int seg = segS[ri];
            float* s = &sums[(size_t)seg * D_IN + col4];
            atomicAdd(s + 0, acc.x);
            atomicAdd(s + 1, acc.y);
            atomicAdd(s + 2, acc.z);
            atomicAdd(s + 3, acc.w);
            if ((t & 63) == 0) atomicAdd(&counts[seg], (float)cnt);
            acc = make_float4(0.f, 0.f, 0.f, 0.f);
            cnt = 0;
        }
    }
}

// -------- kernel 2: fused mean-div + GEMM(WMMA f32) + LN + head ------------
// One block per 16 pooled rows. 8 waves; wave w computes the 16x16 tile of
// h = pooled @ W1 for columns [16w, 16w+16) with 64 chained
// v_wmma_f32_16x16x4_f32 over K=256.
__global__ void __launch_bounds__(256)
head_kernel(const float* __restrict__ sums, const float* __restrict__ counts,
            const float* __restrict__ W1, const float* __restrict__ b1,
            const float* __restrict__ gamma, const float* __restrict__ beta,
            const float* __restrict__ W2, const float* __restrict__ b2,
            float* __restrict__ out) {
    __shared__ float poolS[16 * POOL_STRIDE];   // pooled tile [16][256], skewed
    __shared__ float hS[16 * H_STRIDE];         // h tile [16][128], skewed

    const int tid  = threadIdx.x;
    const int m0   = blockIdx.x * 16;
    const int wave = tid >> 5;                  // 0..7 -> N tile
    const int lane = tid & 31;
    const int l15  = lane & 15;

    // Stage pooled tile: divide sums by counts on the way in.
    for (int r = 0; r < 16; ++r) {
        const float c   = counts[m0 + r];
        const float inv = 1.0f / fmaxf(c, 1.0f);
        poolS[r * POOL_STRIDE + tid] = sums[(size_t)(m0 + r) * D_IN + tid] * inv;
    }
    __syncthreads();

    // A-matrix (16x4 f32) lane layout: lanes 0-15 -> K offsets {0,1},
    // lanes 16-31 -> K offsets {2,3}; M = lane & 15.  B (4x16) mirrors it.
    const int khalf = (lane >> 4) << 1;         // 0 or 2
    const int ncol  = wave * 16 + l15;          // global column of h

    v8f c = {};
    for (int kk = 0; kk < D_IN / 4; ++kk) {
        const int kb = kk * 4 + khalf;
        const v2f a = *(const v2f*)&poolS[l15 * POOL_STRIDE + kb];
        v2f b;
        b.x = W1[(size_t)kb * D_H + ncol];
        b.y = W1[(size_t)(kb + 1) * D_H + ncol];
        c = __builtin_amdgcn_wmma_f32_16x16x4_f32(
                /*neg_a=*/false, a, /*neg_b=*/false, b,
                /*c_mod=*/(short)0, c, /*reuse_a=*/false, /*reuse_b=*/false);
    }

    // C/D layout: VGPR v, lanes 0-15 -> M=v, lanes 16-31 -> M=v+8; N = lane&15.
    const int mhi = (lane >> 4) << 3;
    const float bias = b1[ncol];
#pragma unroll
    for (int v = 0; v < 8; ++v) {
        hS[(v + mhi) * H_STRIDE + ncol] = c[v] + bias;
    }
    __syncthreads();

    // LayerNorm + LeakyReLU + 128->1 dot: 16 rows, one thread each (tiny).
    if (tid < 16) {
        const int row = tid;
        float s = 0.0f, s2 = 0.0f;
        for (int n = 0; n < D_H; ++n) {
            const float v = hS[row * H_STRIDE + n];
            s  += v;
            s2 += v * v;
        }
        const float mu  = s * (1.0f / D_H);
        const float var = s2 * (1.0f / D_H) - mu * mu;
        const float rs  = rsqrtf(var + LN_EPS);
        float acc = 0.0f;
        for (int n = 0; n < D_H; ++n) {
            float v = (hS[row * H_STRIDE + n] - mu) * rs * gamma[n] + beta[n];
            v = (v >= 0.0f) ? v : NEG_SLOPE * v;
            acc += v * W2[n];
        }
        out[m0 + row] = acc + b2[0];
    }
}

// ---------------------------------------------------------------------------
extern "C" void kernel_launch(void* const* d_in, const int* in_sizes, int n_in,
                              void* d_out, int out_size, void* d_ws, size_t ws_size,
                              hipStream_t stream) {
    const float* features = (const float*)d_in[0];   // [500000, 256]
    const int*   batch    = (const int*)d_in[1];     // [500000] sorted segment ids
    const float* W1       = (const float*)d_in[2];   // [256, 128]
    const float* b1       = (const float*)d_in[3];   // [128]
    const float* gamma    = (const float*)d_in[4];   // [128]
    const float* beta     = (const float*)d_in[5];   // [128]
    const float* W2       = (const float*)d_in[6];   // [128, 1]
    const float* b2       = (const float*)d_in[7];   // [1]
    float* out = (float*)d_out;                      // [2048]

    const int n_rows = in_sizes[1];                  // 500000

    // Workspace layout: sums [2048*256] f32, then counts [2048] f32.
    float* sums   = (float*)d_ws;
    float* counts = sums + (size_t)NUM_GRAPHS * D_IN;
    const int n_zero = NUM_GRAPHS * D_IN + NUM_GRAPHS;

    zero_ws_kernel<<<(n_zero + 255) / 256, 256, 0, stream>>>(sums, n_zero);

    const int n_blocks = (n_rows + ROWS_PER_BLK - 1) / ROWS_PER_BLK;
    seg_sum_kernel<<<n_blocks, 256, 0, stream>>>(features, batch, sums, counts, n_rows);

    head_kernel<<<NUM_GRAPHS / 16, 256, 0, stream>>>(sums, counts, W1, b1, gamma, beta,
                                                     W2, b2, out);
}